// ArithmeticCircuitGNN_89730456748625
// MI455X (gfx1250) — compile-verified
//
#include <hip/hip_runtime.h>
#include <hip/hip_bf16.h>

// ---------------------------------------------------------------------------
// GCN (4 layers, residual + LayerNorm) for gfx1250 / MI455X.
// GEMMs: V_WMMA_F32_16X16X4_F32 (full fp32; problem is scatter/memory bound,
// so low-precision WMMA buys nothing end-to-end) with the weight tile staged
// into LDS via GLOBAL_LOAD_ASYNC_TO_LDS_B128 (+ s_wait_asynccnt) and consumed
// through ds_load at immediate offsets -> no 64-bit address math in the loop.
// ---------------------------------------------------------------------------

typedef __attribute__((ext_vector_type(2))) float v2f;
typedef __attribute__((ext_vector_type(8))) float v8f;
typedef int v4i_t __attribute__((vector_size(16)));
typedef __attribute__((address_space(1))) v4i_t* gptr_v4i;
typedef __attribute__((address_space(3))) v4i_t* lptr_v4i;

#define EPS_LN 1e-5f
#define TILE_N 64   // columns of W staged per workgroup
#define KC     128  // K-chunk staged per barrier round (32 KB LDS)

#if defined(__has_builtin)
#if __has_builtin(__builtin_amdgcn_global_load_async_to_lds_b128)
#define HAVE_ASYNC_LDS 1
#endif
#endif

// ----------------------------- tiny utility kernels ------------------------

__global__ void fill_kernel(float* __restrict__ p, float v, size_t n) {
    size_t i = (size_t)blockIdx.x * blockDim.x + threadIdx.x;
    if (i < n) p[i] = v;
}

__global__ void deg_accum_kernel(const long long* __restrict__ dst,
                                 float* __restrict__ deg, int E) {
    int e = blockIdx.x * blockDim.x + threadIdx.x;
    if (e < E) atomicAdd(&deg[dst[e]], 1.0f);
}

__global__ void rsqrt_inplace_kernel(float* __restrict__ p, int n) {
    int i = blockIdx.x * blockDim.x + threadIdx.x;
    if (i < n) p[i] = rsqrtf(p[i]);
}

// ----------------------------- WMMA fp32 GEMM ------------------------------
// C[N, Dout] = A[N, K] * W[K, Dout].
// Requires: N % 32 == 0, Dout % 64 == 0, K % KC == 0 (true here:
// N = 100000, K in {128,256}, Dout in {256,128}).
// Workgroup: 256 threads = 8 waves arranged 2(M) x 4(N) -> 32x64 output tile.
__global__ __launch_bounds__(256) void gemm_f32_wmma(
    const float* __restrict__ A, const float* __restrict__ W,
    float* __restrict__ C, int K, int Dout) {
    __shared__ float Bs[KC * TILE_N];  // 32 KB

    const int tid   = threadIdx.x;
    const int wave  = tid >> 5;
    const int lane  = tid & 31;
    const int wm    = wave & 1;          // M tile within workgroup
    const int wn    = wave >> 1;         // N tile within workgroup
    const int m0    = blockIdx.x * 32 + wm * 16;
    const int nbase = blockIdx.y * TILE_N;
    const int half  = lane >> 4;         // 0: lanes 0-15, 1: lanes 16-31
    const int lid   = lane & 15;

    // A fragment source: lane (half,lid) holds A[m0+lid][k + 2*half + {0,1}]
    const float* arow = A + (size_t)(m0 + lid) * K + 2 * half;
    // B fragment source (LDS): VGPR r holds Bs[(kk + 2*half + r)*64 + wn*16 + lid]
    const float* bsp = Bs + (2 * half) * TILE_N + wn * 16 + lid;

    v8f acc = {};
    for (int k0 = 0; k0 < K; k0 += KC) {
        // ---- stage W[k0:k0+KC, nbase:nbase+64] into LDS ------------------
        // KC*TILE_N = 8192 floats; 256 threads x 4 floats -> 8 rounds; all
        // loop trip counts are workgroup-uniform (EXEC stays all-ones).
#pragma unroll
        for (int c = tid * 4; c < KC * TILE_N; c += 256 * 4) {
            const int row = c >> 6;   // /64
            const int col = c & 63;
            const float* gsrc = W + (size_t)(k0 + row) * Dout + nbase + col;
#if defined(HAVE_ASYNC_LDS)
            __builtin_amdgcn_global_load_async_to_lds_b128(
                (gptr_v4i)(uintptr_t)gsrc,
                (lptr_v4i)(unsigned)(uintptr_t)(&Bs[c]),
                0, 0);
#else
            *(float4*)(&Bs[c]) = *(const float4*)gsrc;
#endif
        }
#if defined(HAVE_ASYNC_LDS)
        asm volatile("s_wait_asynccnt 0x0" ::: "memory");
#endif
        __syncthreads();

        // ---- consume chunk: 32 WMMA steps of K=4 -------------------------
#pragma unroll 8
        for (int kk = 0; kk < KC; kk += 4) {
            const int k = k0 + kk;
            v2f a = { arow[k], arow[k + 1] };
            v2f b = { bsp[kk * TILE_N], bsp[(kk + 1) * TILE_N] };
            // (neg_a, A, neg_b, B, c_mod, C, reuse_a, reuse_b)
            acc = __builtin_amdgcn_wmma_f32_16x16x4_f32(
                false, a, false, b, (short)0, acc, false, false);
        }
        __syncthreads();
    }

    // C/D layout: VGPR r -> row m0 + r + 8*half, col nbase + wn*16 + lid
    float* crow = C + (size_t)(m0 + half * 8) * Dout + nbase + wn * 16 + lid;
#pragma unroll
    for (int r = 0; r < 8; ++r) crow[(size_t)r * Dout] = acc[r];
}

// ----------------------------- edge scatter --------------------------------
// One block per edge: agg[dst] += P[src] * dinv[src] * dinv[dst].
// src/dst loads are block-uniform -> scalar loads; fp32 global atomics.
__global__ void scatter_kernel(const long long* __restrict__ src,
                               const long long* __restrict__ dst,
                               const float* __restrict__ P,
                               const float* __restrict__ dinv,
                               float* __restrict__ agg, int D, int E) {
    int e = blockIdx.x;
    if (e >= E) return;
    long long s = src[e];
    long long d = dst[e];
    float w = dinv[s] * dinv[d];
    size_t sb = (size_t)s * D;
    size_t db = (size_t)d * D;
    for (int f = threadIdx.x; f < D; f += blockDim.x)
        atomicAdd(&agg[db + f], P[sb + f] * w);
}

// ----------------------------- fused post + LN -----------------------------
// One wave (32 lanes) per node row; D/32 features per lane (4 or 8).
//   u = agg[row] + P[row]*dinv[row]^2 + bias          (self-loop folded in)
//   if relu: u = max(u,0); if hres: u += hres[row]    (residual)
//   if hout: hout[row] = u; if tout: tout[row] = LayerNorm(u)*g + b
// agg/tout and hres/hout may alias (elementwise per-row -> safe).
__global__ __launch_bounds__(256) void post_ln_kernel(
    const float* agg, const float* __restrict__ P,
    const float* __restrict__ dinv, const float* __restrict__ bias,
    const float* hres, float* hout,
    const float* __restrict__ g, const float* __restrict__ bln,
    float* tout, int n, int D, int relu_flag) {
    const int wid  = threadIdx.x >> 5;
    const int lane = threadIdx.x & 31;
    const int row  = blockIdx.x * (blockDim.x >> 5) + wid;
    if (row >= n) return;

    const int nf = D >> 5;  // 4 (D=128) or 8 (D=256)
    float v[8];
    const float dv  = dinv[row];
    const float dv2 = dv * dv;
    const size_t base = (size_t)row * D;

    for (int j = 0; j < nf; ++j) {
        int f = (j << 5) + lane;
        float u = agg[base + f] + P[base + f] * dv2 + bias[f];
        if (relu_flag) u = fmaxf(u, 0.0f);
        if (hres) u += hres[base + f];
        v[j] = u;
        if (hout) hout[base + f] = u;
    }
    if (!tout) return;

    // wave32 LayerNorm
    float s = 0.0f;
    for (int j = 0; j < nf; ++j) s += v[j];
    for (int o = 16; o > 0; o >>= 1) s += __shfl_xor(s, o, 32);
    float mu = s / (float)D;

    float q = 0.0f;
    for (int j = 0; j < nf; ++j) { float d = v[j] - mu; q += d * d; }
    for (int o = 16; o > 0; o >>= 1) q += __shfl_xor(q, o, 32);
    float inv = rsqrtf(q / (float)D + EPS_LN);

    for (int j = 0; j < nf; ++j) {
        int f = (j << 5) + lane;
        tout[base + f] = (v[j] - mu) * inv * g[f] + bln[f];
    }
}

// ----------------------------- host orchestration --------------------------

extern "C" void kernel_launch(void* const* d_in, const int* in_sizes, int n_in,
                              void* d_out, int out_size, void* d_ws, size_t ws_size,
                              hipStream_t stream) {
    const float*     x    = (const float*)d_in[0];
    const long long* ei   = (const long long*)d_in[1];
    const float* W0 = (const float*)d_in[2];  const float* b0 = (const float*)d_in[3];
    const float* W1 = (const float*)d_in[4];  const float* b1 = (const float*)d_in[5];
    const float* W2 = (const float*)d_in[6];  const float* b2 = (const float*)d_in[7];
    const float* W3 = (const float*)d_in[8];  const float* b3 = (const float*)d_in[9];
    const float* ln0g = (const float*)d_in[10]; const float* ln0b = (const float*)d_in[11];
    const float* ln1g = (const float*)d_in[12]; const float* ln1b = (const float*)d_in[13];
    const float* flng = (const float*)d_in[14]; const float* flnb = (const float*)d_in[15];

    const int N = in_sizes[0] / 128;   // 100000
    const int E = in_sizes[1] / 2;     // 300000
    const long long* src = ei;         // edge_index[0]
    const long long* dst = ei + E;     // edge_index[1]

    // Workspace layout: dinv[N] | P[N*256] | agg/t[N*256] | h[N*256]
    char* ws = (char*)d_ws;
    const size_t big    = (size_t)N * 256 * sizeof(float);
    const size_t p_off  = (((size_t)N * sizeof(float)) + 255) & ~(size_t)255;
    float* dinv = (float*)ws;
    float* P    = (float*)(ws + p_off);
    float* aggt = (float*)(ws + p_off + big);        // shared agg / LN-output t
    float* h    = (float*)(ws + p_off + 2 * big);    // residual state
    float* outp = (float*)d_out;

    const int tb = 256;

    // degrees: deg = 1 (self loop) + count over dst; then dinv = rsqrt(deg)
    fill_kernel<<<(N + tb - 1) / tb, tb, 0, stream>>>(dinv, 1.0f, (size_t)N);
    deg_accum_kernel<<<(E + tb - 1) / tb, tb, 0, stream>>>(dst, dinv, E);
    rsqrt_inplace_kernel<<<(N + tb - 1) / tb, tb, 0, stream>>>(dinv, N);

    auto run_layer = [&](const float* Ain, int K, int D,
                         const float* Wp, const float* bp,
                         const float* hres, float* houtp,
                         const float* g, const float* bl, float* tp, int relu) {
        dim3 gg((unsigned)(N / 32), (unsigned)(D / 64));
        gemm_f32_wmma<<<gg, 256, 0, stream>>>(Ain, Wp, P, K, D);
        size_t tot = (size_t)N * D;
        fill_kernel<<<(unsigned)((tot + tb - 1) / tb), tb, 0, stream>>>(aggt, 0.0f, tot);
        scatter_kernel<<<E, D, 0, stream>>>(src, dst, P, dinv, aggt, D, E);
        post_ln_kernel<<<(N + 7) / 8, 256, 0, stream>>>(
            aggt, P, dinv, bp, hres, houtp, g, bl, tp, N, D, relu);
    };

    // layer 0: h0 = relu(gcn(x, W0, b0));              t = ln0(h0)
    run_layer(x,    128, 256, W0, b0, nullptr, h, ln0g, ln0b, aggt, 1);
    // layer 1: h1 = relu(gcn(t, W1, b1)) + h0;         t = ln1(h1)
    run_layer(aggt, 256, 256, W1, b1, h,       h, ln1g, ln1b, aggt, 1);
    // layer 2: h2 = relu(gcn(t, W2, b2)) + h1          (no LN output)
    run_layer(aggt, 256, 256, W2, b2, h,       h, nullptr, nullptr, nullptr, 1);
    // layer 3: out = fln(gcn(h2, W3, b3))              (no relu/residual)
    run_layer(h,    256, 128, W3, b3, nullptr, nullptr, flng, flnb, outp, 0);
}